// PostProcessor_4243427688630
// MI455X (gfx1250) — compile-verified
//
#include <hip/hip_runtime.h>
#include <math.h>

// ---- problem constants (match reference) ----
#define N_PROP 1024
#define NCLS 81
#define NCLS_FG 80
#define FDIM 1024
#define KDET 100
#define IMG_W 1216.0f
#define IMG_H 800.0f
#define SCORE_THRESH 0.05f
#define NMS_THRESH 0.5f
#define BBOX_CLIP 4.135166556742356f   // log(1000/16)
#define NEG_BIG -3.0e38f

typedef __attribute__((ext_vector_type(2))) float v2f;
typedef __attribute__((ext_vector_type(8))) float v8f;

#if defined(__HIP_DEVICE_COMPILE__) && defined(__gfx1250__) && __has_builtin(__builtin_amdgcn_wmma_f32_16x16x4_f32)
#define HAVE_WMMA_F32X4 1
#else
#define HAVE_WMMA_F32X4 0
#endif

// ------------------------------------------------------------------
// Kernel A: per-proposal softmax + box decode + clip.
// grid = N_PROP blocks, block = 128 threads (thread c handles class c).
// Outputs class-major: probs_cls[c][n], boxes_cls[(c*N+n)*4+j] for fg c=0..79.
// ------------------------------------------------------------------
__global__ void k_decode_softmax(const float* __restrict__ logits,
                                 const float* __restrict__ boxreg,
                                 const float* __restrict__ props,
                                 float* __restrict__ probs_cls,
                                 float* __restrict__ boxes_cls) {
  const int n = blockIdx.x;
  const int tid = threadIdx.x;
  __shared__ float red[128];

  float l = (tid < NCLS) ? logits[n * NCLS + tid] : NEG_BIG;
  red[tid] = l;
  __syncthreads();
  for (int s = 64; s > 0; s >>= 1) {
    if (tid < s) red[tid] = fmaxf(red[tid], red[tid + s]);
    __syncthreads();
  }
  const float mx = red[0];
  __syncthreads();

  float e = (tid < NCLS) ? expf(l - mx) : 0.0f;
  red[tid] = e;
  __syncthreads();
  for (int s = 64; s > 0; s >>= 1) {
    if (tid < s) red[tid] += red[tid + s];
    __syncthreads();
  }
  const float denom = red[0];

  if (tid >= 1 && tid < NCLS) {
    const int c = tid - 1;                       // foreground class index
    probs_cls[c * N_PROP + n] = e / denom;

    const float x1 = props[n * 4 + 0], y1 = props[n * 4 + 1];
    const float x2 = props[n * 4 + 2], y2 = props[n * 4 + 3];
    const float w = x2 - x1 + 1.0f, h = y2 - y1 + 1.0f;
    const float cx = x1 + 0.5f * w, cy = y1 + 0.5f * h;

    const float* rel = &boxreg[n * NCLS * 4 + tid * 4];
    const float dx = rel[0] * 0.1f;                          // /WX
    const float dy = rel[1] * 0.1f;                          // /WY
    const float dw = fminf(rel[2] * 0.2f, BBOX_CLIP);        // /WW, clamped
    const float dh = fminf(rel[3] * 0.2f, BBOX_CLIP);        // /WH, clamped
    const float pcx = dx * w + cx, pcy = dy * h + cy;
    const float pw = expf(dw) * w, ph = expf(dh) * h;

    float ox1 = pcx - 0.5f * pw;
    float oy1 = pcy - 0.5f * ph;
    float ox2 = pcx + 0.5f * pw - 1.0f;
    float oy2 = pcy + 0.5f * ph - 1.0f;
    ox1 = fminf(fmaxf(ox1, 0.0f), IMG_W - 1.0f);
    oy1 = fminf(fmaxf(oy1, 0.0f), IMG_H - 1.0f);
    ox2 = fminf(fmaxf(ox2, 0.0f), IMG_W - 1.0f);
    oy2 = fminf(fmaxf(oy2, 0.0f), IMG_H - 1.0f);

    float* bo = &boxes_cls[(size_t)(c * N_PROP + n) * 4];
    bo[0] = ox1; bo[1] = oy1; bo[2] = ox2; bo[3] = oy2;
  }
}

// ------------------------------------------------------------------
// Kernel B: per-class NMS. grid = 80 blocks, block = 1024 threads.
// Bitonic sort (descending by score, invalid -> -inf key) in LDS, then
// the sequential suppression loop with keep flags in LDS (one barrier
// per i, matching the reference fori_loop exactly).
// ------------------------------------------------------------------
__global__ void k_nms(const float* __restrict__ probs_cls,
                      const float* __restrict__ boxes_cls,
                      float* __restrict__ bs_sorted,
                      int* __restrict__ order_sorted,
                      float* __restrict__ flat_s) {
  const int c = blockIdx.x;
  const int tid = threadIdx.x;

  __shared__ float skey[N_PROP];
  __shared__ int   sidx[N_PROP];
  __shared__ float bx1[N_PROP], by1[N_PROP], bx2[N_PROP], by2[N_PROP], ar[N_PROP];
  __shared__ unsigned char keep[N_PROP];

  const float sc0 = probs_cls[c * N_PROP + tid];
  skey[tid] = (sc0 > SCORE_THRESH) ? sc0 : NEG_BIG;
  sidx[tid] = tid;
  __syncthreads();

  // bitonic sort, descending
  for (int k = 2; k <= N_PROP; k <<= 1) {
    for (int j = k >> 1; j > 0; j >>= 1) {
      const int ixj = tid ^ j;
      if (ixj > tid) {
        const bool desc = ((tid & k) == 0);
        const float a = skey[tid], b = skey[ixj];
        const bool sw = desc ? (a < b) : (a > b);
        if (sw) {
          skey[tid] = b; skey[ixj] = a;
          const int t = sidx[tid]; sidx[tid] = sidx[ixj]; sidx[ixj] = t;
        }
      }
      __syncthreads();
    }
  }

  const int oi = sidx[tid];
  const bool v = skey[tid] > -1.0e38f;
  const float s_act = probs_cls[c * N_PROP + oi];
  const float* bp = &boxes_cls[(size_t)(c * N_PROP + oi) * 4];
  const float x1 = bp[0], y1 = bp[1], x2 = bp[2], y2 = bp[3];
  const float a = (x2 - x1 + 1.0f) * (y2 - y1 + 1.0f);
  bx1[tid] = x1; by1[tid] = y1; bx2[tid] = x2; by2[tid] = y2; ar[tid] = a;
  keep[tid] = v ? 1 : 0;
  order_sorted[c * N_PROP + tid] = oi;
  float* bso = &bs_sorted[(size_t)(c * N_PROP + tid) * 4];
  bso[0] = x1; bso[1] = y1; bso[2] = x2; bso[3] = y2;

  // sequential suppression: kept box i suppresses j>i with IoU > thresh
  for (int i = 0; i < N_PROP - 1; ++i) {
    __syncthreads();
    if (keep[i] && tid > i && keep[tid]) {
      const float ix1 = fmaxf(x1, bx1[i]);
      const float iy1 = fmaxf(y1, by1[i]);
      const float ix2 = fminf(x2, bx2[i]);
      const float iy2 = fminf(y2, by2[i]);
      const float iw = fmaxf(ix2 - ix1 + 1.0f, 0.0f);
      const float ih = fmaxf(iy2 - iy1 + 1.0f, 0.0f);
      const float inter = iw * ih;
      const float iou = inter / (a + ar[i] - inter);
      if (iou > NMS_THRESH) keep[tid] = 0;
    }
  }
  __syncthreads();

  flat_s[c * N_PROP + tid] = keep[tid] ? s_act : -1.0f;
}

// ------------------------------------------------------------------
// Kernel C: top-100 over 80*1024 scores (single block, 1024 threads;
// thread t owns stride slice {t, 1024+t, ...} so "clear selected" needs
// no cross-thread memory visibility), then write boxes/scores/labels
// and the masked original-proposal indices for the feature gather.
// ------------------------------------------------------------------
__global__ void k_topk_finalize(float* __restrict__ flat_s,
                                const float* __restrict__ bs_sorted,
                                const int* __restrict__ order_sorted,
                                float* __restrict__ out,          // d_out base
                                int* __restrict__ orig_masked) {
  const int tid = threadIdx.x;
  __shared__ float rv[N_PROP];
  __shared__ int   ri[N_PROP];
  __shared__ float tops[KDET];
  __shared__ int   topi[KDET];

  for (int k = 0; k < KDET; ++k) {
    float bv = NEG_BIG;
    int bi = 0x3fffffff;
    for (int j = 0; j < NCLS_FG; ++j) {
      const int idx = j * N_PROP + tid;
      const float vv = flat_s[idx];
      if (vv > bv) { bv = vv; bi = idx; }   // first (lowest idx) wins ties
    }
    rv[tid] = bv; ri[tid] = bi;
    __syncthreads();
    for (int s = 512; s > 0; s >>= 1) {
      if (tid < s) {
        const float ov = rv[tid + s];
        const int oi2 = ri[tid + s];
        if (ov > rv[tid] || (ov == rv[tid] && oi2 < ri[tid])) {
          rv[tid] = ov; ri[tid] = oi2;
        }
      }
      __syncthreads();
    }
    const int sel = ri[0];
    if (tid == 0) { topi[k] = sel; tops[k] = rv[0]; }
    if ((sel & (N_PROP - 1)) == tid) flat_s[sel] = NEG_BIG;  // owner-only clear
    __syncthreads();
  }

  if (tid < KDET) {
    const int fi = topi[tid];
    const float sc = tops[tid];
    const bool ok = sc > 0.0f;
    const int cls = fi >> 10;                 // / N_PROP
    const float* bp = &bs_sorted[(size_t)fi * 4];
    out[tid * 4 + 0] = ok ? bp[0] : 0.0f;     // boxes_out [100,4]
    out[tid * 4 + 1] = ok ? bp[1] : 0.0f;
    out[tid * 4 + 2] = ok ? bp[2] : 0.0f;
    out[tid * 4 + 3] = ok ? bp[3] : 0.0f;
    out[4 * KDET + tid] = ok ? sc : 0.0f;                       // scores_out
    out[4 * KDET + KDET + (size_t)KDET * FDIM + tid] =
        ok ? (float)(cls + 1) : 0.0f;                           // labels (as f32)
    orig_masked[tid] = ok ? order_sorted[fi] : -1;
  }
}

// ------------------------------------------------------------------
// Kernel D: feature gather as one-hot f32 GEMM on the WMMA pipe.
// feat_out[100,1024] = OneHot[100,1024] x features[1024,1024]  (exact in f32)
// grid = 7 M-tiles * 64 F-tiles, one wave (32 threads) per 16x16 tile.
// V_WMMA_F32_16X16X4_F32 lane layout:
//   A (16x4): lanes 0-15 -> (M=lane, K=0/1), lanes 16-31 -> (M=lane-16, K=2/3)
//   B (4x16): lanes 0-15 -> (K=0/1, N=lane), lanes 16-31 -> (K=2/3, N=lane-16)
//   C/D v8f:  vgpr r -> M = r + (lane<16 ? 0 : 8), N = lane&15
// ------------------------------------------------------------------
__global__ void k_gather_feat(const float* __restrict__ features,
                              const int* __restrict__ orig_masked,
                              float* __restrict__ feat_out) {
  const int lane = threadIdx.x;        // 0..31
  const int mt = blockIdx.x >> 6;      // 0..6
  const int ft = blockIdx.x & 63;      // 0..63
  const int m0 = mt * 16, f0 = ft * 16;
  const int nl = lane & 15;
  const int mrow = m0 + nl;
  const int my_orig = (mrow < KDET) ? orig_masked[mrow] : -1;
  const int khalf = (lane < 16) ? 0 : 2;

#if HAVE_WMMA_F32X4
#if __has_builtin(__builtin_amdgcn_s_wait_tensorcnt)
  __builtin_amdgcn_s_wait_tensorcnt(0);
#endif
  v8f acc = {};
  for (int kb = 0; kb < N_PROP; kb += 4) {
    v2f a, b;
    a.x = (my_orig == kb + khalf) ? 1.0f : 0.0f;
    a.y = (my_orig == kb + khalf + 1) ? 1.0f : 0.0f;
    b.x = features[(size_t)(kb + khalf) * FDIM + f0 + nl];
    b.y = features[(size_t)(kb + khalf + 1) * FDIM + f0 + nl];
    if (kb + 8 < N_PROP)
      __builtin_prefetch(&features[(size_t)(kb + 8) * FDIM + f0], 0, 1);
    acc = __builtin_amdgcn_wmma_f32_16x16x4_f32(
        /*neg_a=*/false, a, /*neg_b=*/false, b,
        /*c_mod=*/(short)0, acc, /*reuse_a=*/false, /*reuse_b=*/false);
  }
  for (int r = 0; r < 8; ++r) {
    const int m = m0 + r + ((lane < 16) ? 0 : 8);
    if (m < KDET) feat_out[(size_t)m * FDIM + f0 + nl] = acc[r];
  }
#else
  // Fallback: direct gather (keeps the build green if the builtin is absent).
  for (int r = 0; r < 8; ++r) {
    const int m = m0 + r + ((lane < 16) ? 0 : 8);
    if (m < KDET) {
      const int o = orig_masked[m];
      feat_out[(size_t)m * FDIM + f0 + nl] =
          (o >= 0) ? features[(size_t)o * FDIM + f0 + nl] : 0.0f;
    }
  }
#endif
}

// ------------------------------------------------------------------
// Host launcher
// ------------------------------------------------------------------
extern "C" void kernel_launch(void* const* d_in, const int* in_sizes, int n_in,
                              void* d_out, int out_size, void* d_ws, size_t ws_size,
                              hipStream_t stream) {
  (void)in_sizes; (void)n_in; (void)out_size; (void)ws_size;
  const float* logits = (const float*)d_in[0];   // [1024,81]
  const float* boxreg = (const float*)d_in[1];   // [1024,324]
  const float* props  = (const float*)d_in[2];   // [1024,4]
  const float* feats  = (const float*)d_in[3];   // [1024,1024]
  float* out = (float*)d_out;                    // 400 | 100 | 102400 | 100

  char* ws = (char*)d_ws;
  size_t off = 0;
  float* probs_cls   = (float*)(ws + off); off += (size_t)NCLS_FG * N_PROP * 4;
  float* boxes_cls   = (float*)(ws + off); off += (size_t)NCLS_FG * N_PROP * 4 * 4;
  float* bs_sorted   = (float*)(ws + off); off += (size_t)NCLS_FG * N_PROP * 4 * 4;
  int*   order_srt   = (int*)  (ws + off); off += (size_t)NCLS_FG * N_PROP * 4;
  float* flat_s      = (float*)(ws + off); off += (size_t)NCLS_FG * N_PROP * 4;
  int*   orig_masked = (int*)  (ws + off); off += 128 * 4;

  k_decode_softmax<<<N_PROP, 128, 0, stream>>>(logits, boxreg, props,
                                               probs_cls, boxes_cls);
  k_nms<<<NCLS_FG, N_PROP, 0, stream>>>(probs_cls, boxes_cls,
                                        bs_sorted, order_srt, flat_s);
  k_topk_finalize<<<1, N_PROP, 0, stream>>>(flat_s, bs_sorted, order_srt,
                                            out, orig_masked);
  // feat_out starts after boxes(400) + scores(100)
  k_gather_feat<<<(KDET + 15) / 16 * 64, 32, 0, stream>>>(feats, orig_masked,
                                                          out + 4 * KDET + KDET);
}